// SpikingNeuron_30751965839549
// MI455X (gfx1250) — compile-verified
//
#include <hip/hip_runtime.h>
#include <stdint.h>

// PLIF spiking-neuron forward scan for x:[B=16,T=32,C=128,H=32,W=32] f32.
// Memory-bound: 512 MB total traffic ~= 22us at 23.3 TB/s. Strategy:
//  - one thread per float4 chain, scan T=32 in registers
//  - CDNA5 async global->LDS b128 prefetch pipeline, depth 3 (ASYNCcnt),
//    to hold ~3x more load bytes in flight (latency-BW product ~20 MB)
//  - nontemporal hints both directions (streaming, no L2 reuse)

typedef float v4f __attribute__((ext_vector_type(4)));

#define T_STEPS 32
#define CHW4 32768u                     // (128*32*32)/4 float4 slots per (b,t) slice
#define TSTRIDE_BYTES (CHW4 * 16u)      // 512 KB between consecutive timesteps
#define DEPTH 3                         // prefetch depth
#define NBUF 4                          // LDS ring buffers (power of two)

__device__ __forceinline__ uint32_t lds_off(const void* p) {
  // addrspace(3) -> generic lowering places the LDS byte offset in the low 32 bits
  return (uint32_t)(uintptr_t)p;
}

__device__ __forceinline__ void async_load_b128(uint32_t lds_byte_addr, uint64_t gaddr) {
  asm volatile("global_load_async_to_lds_b128 %0, %1, off th:TH_LOAD_NT"
               :: "v"(lds_byte_addr), "v"(gaddr)
               : "memory");
}

__device__ __forceinline__ void wait_async(int n) {
  // n is compile-time constant after full unroll
  switch (n) {
    case 0:  asm volatile("s_wait_asynccnt 0x0" ::: "memory"); break;
    case 1:  asm volatile("s_wait_asynccnt 0x1" ::: "memory"); break;
    case 2:  asm volatile("s_wait_asynccnt 0x2" ::: "memory"); break;
    default: asm volatile("s_wait_asynccnt 0x3" ::: "memory"); break;
  }
}

__global__ __launch_bounds__(256) void plif_scan_kernel(
    const float* __restrict__ x, const float* __restrict__ a_ptr,
    float* __restrict__ out) {
  __shared__ v4f buf[NBUF][256];        // 16 KB: per-lane private ring-buffer slots
  const int tid = threadIdx.x;
  const uint32_t f    = blockIdx.x * 256u + (uint32_t)tid;  // float4 chain id
  const uint32_t b    = f / CHW4;
  const uint32_t chw4 = f % CHW4;

  const float decay = 1.0f / (1.0f + expf(-a_ptr[0]));      // sigmoid(a)

  const uint64_t gbase =
      (uint64_t)(uintptr_t)x +
      (uint64_t)b * (uint64_t)T_STEPS * (uint64_t)TSTRIDE_BYTES +
      (uint64_t)chw4 * 16ull;

  v4f* out4 = (v4f*)out;
  const uint64_t obase = (uint64_t)b * T_STEPS * CHW4 + chw4;

  uint32_t l[NBUF];
#pragma unroll
  for (int i = 0; i < NBUF; ++i) l[i] = lds_off(&buf[i][tid]);

  // prime the pipeline: t = 0 .. DEPTH-1
#pragma unroll
  for (int t = 0; t < DEPTH; ++t)
    async_load_b128(l[t], gbase + (uint64_t)t * (uint64_t)TSTRIDE_BYTES);

  v4f mem = {0.0f, 0.0f, 0.0f, 0.0f};

#pragma unroll
  for (int t = 0; t < T_STEPS; ++t) {
    if (t + DEPTH < T_STEPS) {
      async_load_b128(l[(t + DEPTH) & (NBUF - 1)],
                      gbase + (uint64_t)(t + DEPTH) * (uint64_t)TSTRIDE_BYTES);
    }
    // ensure the oldest outstanding load (timestep t) has landed in LDS
    const int remaining = (T_STEPS - 1 - t) < DEPTH ? (T_STEPS - 1 - t) : DEPTH;
    wait_async(remaining);

    const v4f xi = buf[t & (NBUF - 1)][tid];  // ds_load_b128, own slot, no barrier

    mem = decay * mem + xi;             // leaky integrate
    v4f s;
#pragma unroll
    for (int k = 0; k < 4; ++k) {
      const bool fire = mem[k] >= 1.0f; // spike_fn(mem - THRESHOLD) with thr = 1
      s[k]   = fire ? 1.0f : 0.0f;
      mem[k] = fire ? 0.0f : mem[k];    // hard reset
    }

    __builtin_nontemporal_store(s, out4 + obase + (uint64_t)t * CHW4);
  }
}

extern "C" void kernel_launch(void* const* d_in, const int* in_sizes, int n_in,
                              void* d_out, int out_size, void* d_ws, size_t ws_size,
                              hipStream_t stream) {
  const float* x = (const float*)d_in[0];
  const float* a = (const float*)d_in[1];
  float* out = (float*)d_out;

  const int total   = in_sizes[0];            // 16*32*128*32*32 = 67,108,864
  const int chains4 = total / (T_STEPS * 4);  // 524,288 float4 chains
  const int blocks  = chains4 / 256;          // 2,048 blocks x 256 threads

  plif_scan_kernel<<<blocks, 256, 0, stream>>>(x, a, out);
}